// MyHGTLayer_60241211293942
// MI455X (gfx1250) — compile-verified
//
#include <hip/hip_runtime.h>
#include <math.h>

#define D 64
#define TNUM 8
#define RNUM 8

typedef __attribute__((ext_vector_type(2))) float v2f;
typedef __attribute__((ext_vector_type(8))) float v8f;

__device__ __forceinline__ v8f wmma_f32(v2f a, v2f b, v8f c) {
    // V_WMMA_F32_16X16X4_F32 : D = A(16x4) * B(4x16) + C(16x16), all f32
    return __builtin_amdgcn_wmma_f32_16x16x4_f32(
        false, a, false, b, (short)0, c, false, false);
}

// order-preserving float <-> uint mapping for atomicMax on possibly-negative floats
__device__ __forceinline__ unsigned fmap(float x) {
    unsigned u = __float_as_uint(x);
    return (u & 0x80000000u) ? ~u : (u | 0x80000000u);
}
__device__ __forceinline__ float funmap(unsigned m) {
    unsigned u = (m & 0x80000000u) ? (m & 0x7FFFFFFFu) : ~m;
    return __uint_as_float(u);
}

// map flat tile id -> (segment t, base element, rows in this tile) over bucketed lists
__device__ __forceinline__ bool find_tile(const int* __restrict__ off, int K, int j,
                                          int& t, int& base, int& cnt) {
    for (int tt = 0; tt < K; ++tt) {
        int c  = off[tt + 1] - off[tt];
        int nt = (c + 15) >> 4;
        if (j < nt) {
            t    = tt;
            base = off[tt] + j * 16;
            int rem = c - j * 16;
            cnt  = rem > 16 ? 16 : rem;   // cnt >= 1 whenever we return true
            return true;
        }
        j -= nt;
    }
    return false;
}

__global__ void k_zero(float* __restrict__ p, long n) {
    long i = (long)blockIdx.x * blockDim.x + threadIdx.x;
    if (i < n) p[i] = 0.0f;
}

__global__ void k_count(const int* __restrict__ types, int n, int* __restrict__ cnt) {
    int i = blockIdx.x * blockDim.x + threadIdx.x;
    if (i < n) atomicAdd(&cnt[types[i]], 1);
}

__global__ void k_scan(const int* __restrict__ cnt, int* __restrict__ off,
                       int* __restrict__ cur, int K) {
    if (threadIdx.x == 0) {
        int acc = 0;
        for (int t = 0; t < K; ++t) { off[t] = acc; cur[t] = acc; acc += cnt[t]; }
        off[K] = acc;
    }
}

__global__ void k_scatter(const int* __restrict__ types, int n,
                          int* __restrict__ cur, int* __restrict__ bucket) {
    int i = blockIdx.x * blockDim.x + threadIdx.x;
    if (i < n) {
        int p = atomicAdd(&cur[types[i]], 1);
        bucket[p] = i;
    }
}

// ---- node-typed projections: kn/qn/vn[row] = h[row] @ {K,Q,V}w[type] ----
// Partial tiles: inactive row slots are clamped to the last valid row; the
// duplicated slots compute bitwise-identical outputs, so unconditional stores
// to the (shared) row address are a benign identical-value race.
__global__ void __launch_bounds__(32)
k_node_proj(const float* __restrict__ h,
            const int* __restrict__ bucket, const int* __restrict__ toff,
            const float* __restrict__ kw, const float* __restrict__ qw,
            const float* __restrict__ vw,
            float* __restrict__ kn, float* __restrict__ qn, float* __restrict__ vn) {
    int t, base, cnt;
    if (!find_tile(toff, TNUM, blockIdx.x, t, base, cnt)) return;
    const int lane = threadIdx.x;
    const int m    = lane & 15;
    const int half = lane >> 4;
    const int kb   = half * 2;

    const int rowA = bucket[base + (m < cnt ? m : cnt - 1)];
    const float* pa = h + (size_t)rowA * D + kb;
    v2f a[16];
#pragma unroll
    for (int k = 0; k < 16; ++k)
        a[k] = *reinterpret_cast<const v2f*>(pa + 4 * k);

    int rowS[8];
#pragma unroll
    for (int v = 0; v < 8; ++v) {
        int rr = v + 8 * half;
        rowS[v] = bucket[base + (rr < cnt ? rr : cnt - 1)];
    }

    const float* Ws[3] = { kw + (size_t)t * D * D, qw + (size_t)t * D * D,
                           vw + (size_t)t * D * D };
    float* Os[3] = { kn, qn, vn };
#pragma unroll
    for (int w = 0; w < 3; ++w) {
        const float* W = Ws[w];
        float*       O = Os[w];
        for (int n = 0; n < 4; ++n) {
            v8f c = {};
            const int col = n * 16 + m;
#pragma unroll
            for (int k = 0; k < 16; ++k) {
                v2f b;
                b.x = W[(4 * k + kb) * D + col];
                b.y = W[(4 * k + kb + 1) * D + col];
                c = wmma_f32(a[k], b, c);
            }
#pragma unroll
            for (int v = 0; v < 8; ++v)
                O[(size_t)rowS[v] * D + col] = c[v];
        }
    }
}

// ---- per-edge attention score: s = ((kn[src] @ Att_r) . qn[dst]) * pri_r / 8 ----
__global__ void __launch_bounds__(32)
k_edge_score(const float* __restrict__ kn, const float* __restrict__ qn,
             const int* __restrict__ adj, int E_,
             const int* __restrict__ eb, const int* __restrict__ roff,
             const float* __restrict__ ratt, const float* __restrict__ rpri,
             float* __restrict__ s, unsigned* __restrict__ mmax) {
    int r, base, cnt;
    if (!find_tile(roff, RNUM, blockIdx.x, r, base, cnt)) return;
    const int lane = threadIdx.x;
    const int m    = lane & 15;
    const int half = lane >> 4;
    const int kb   = half * 2;

    const int eA   = eb[base + (m < cnt ? m : cnt - 1)];
    const int srcA = adj[eA];
    const float* pa = kn + (size_t)srcA * D + kb;
    v2f a[16];
#pragma unroll
    for (int k = 0; k < 16; ++k)
        a[k] = *reinterpret_cast<const v2f*>(pa + 4 * k);

    int eR[8], dstR[8];
#pragma unroll
    for (int v = 0; v < 8; ++v) {
        int rr  = v + 8 * half;
        eR[v]   = eb[base + (rr < cnt ? rr : cnt - 1)];
        dstR[v] = adj[E_ + eR[v]];
    }
    const float* A = ratt + (size_t)r * D * D;
    float pd[8] = {0.f, 0.f, 0.f, 0.f, 0.f, 0.f, 0.f, 0.f};
    for (int n = 0; n < 4; ++n) {
        v8f c = {};
        const int col = n * 16 + m;
#pragma unroll
        for (int k = 0; k < 16; ++k) {
            v2f b;
            b.x = A[(4 * k + kb) * D + col];
            b.y = A[(4 * k + kb + 1) * D + col];
            c = wmma_f32(a[k], b, c);
        }
#pragma unroll
        for (int v = 0; v < 8; ++v)
            pd[v] += c[v] * qn[(size_t)dstR[v] * D + col];
    }
    // reduce the row-dot partials across the 16 lanes of each half-wave
#pragma unroll
    for (int off = 1; off < 16; off <<= 1)
#pragma unroll
        for (int v = 0; v < 8; ++v)
            pd[v] += __shfl_xor(pd[v], off, 32);

    if (m == 0) {
        float scale = rpri[r] * 0.125f;   // 1/sqrt(64)
#pragma unroll
        for (int v = 0; v < 8; ++v) {
            float sc = pd[v] * scale;
            s[eR[v]] = sc;                            // dup slots: identical value
            atomicMax(&mmax[dstR[v]], fmap(sc));      // dup slots: idempotent
        }
    }
}

__global__ void k_exp_sum(float* __restrict__ s, const unsigned* __restrict__ mmax,
                          float* __restrict__ ssum, const int* __restrict__ adj, int E_) {
    int e = blockIdx.x * blockDim.x + threadIdx.x;
    if (e < E_) {
        int dv   = adj[E_ + e];
        float ex = expf(s[e] - funmap(mmax[dv]));
        s[e] = ex;
        atomicAdd(&ssum[dv], ex);
    }
}

// ---- messages: agg[dst] += alpha * (vn[src] @ Msg_r) ----
__global__ void __launch_bounds__(32)
k_edge_msg(const float* __restrict__ vn, const int* __restrict__ adj, int E_,
           const int* __restrict__ eb, const int* __restrict__ roff,
           const float* __restrict__ rmsg,
           const float* __restrict__ s, const float* __restrict__ ssum,
           float* agg) {
    int r, base, cnt;
    if (!find_tile(roff, RNUM, blockIdx.x, r, base, cnt)) return;
    const int lane = threadIdx.x;
    const int m    = lane & 15;
    const int half = lane >> 4;
    const int kb   = half * 2;

    const int eA   = eb[base + (m < cnt ? m : cnt - 1)];
    const int srcA = adj[eA];
    const float* pa = vn + (size_t)srcA * D + kb;
    v2f a[16];
#pragma unroll
    for (int k = 0; k < 16; ++k)
        a[k] = *reinterpret_cast<const v2f*>(pa + 4 * k);

    int dstR[8];
    float alpha[8];
#pragma unroll
    for (int v = 0; v < 8; ++v) {
        int rr   = v + 8 * half;
        bool act = (rr < cnt);
        int e2   = eb[base + (act ? rr : cnt - 1)];
        dstR[v]  = adj[E_ + e2];
        // alpha must be 0 for duplicated slots: atomicAdd would double-count
        alpha[v] = act ? s[e2] / (ssum[dstR[v]] + 1e-16f) : 0.f;
    }
    const float* M = rmsg + (size_t)r * D * D;
    for (int n = 0; n < 4; ++n) {
        v8f c = {};
        const int col = n * 16 + m;
#pragma unroll
        for (int k = 0; k < 16; ++k) {
            v2f b;
            b.x = M[(4 * k + kb) * D + col];
            b.y = M[(4 * k + kb + 1) * D + col];
            c = wmma_f32(a[k], b, c);
        }
#pragma unroll
        for (int v = 0; v < 8; ++v)
            atomicAdd(&agg[(size_t)dstR[v] * D + col], c[v] * alpha[v]);
    }
}

// ---- output: io[row] = (io[row] @ Aw[type]) * sigmoid(skip[type])  (in-place safe) ----
__global__ void __launch_bounds__(32)
k_out(const int* __restrict__ bucket, const int* __restrict__ toff,
      const float* __restrict__ aw, const float* __restrict__ skip,
      float* io) {
    int t, base, cnt;
    if (!find_tile(toff, TNUM, blockIdx.x, t, base, cnt)) return;
    const int lane = threadIdx.x;
    const int m    = lane & 15;
    const int half = lane >> 4;
    const int kb   = half * 2;

    const int rowA = bucket[base + (m < cnt ? m : cnt - 1)];
    const float* pa = io + (size_t)rowA * D + kb;
    v2f a[16];
#pragma unroll
    for (int k = 0; k < 16; ++k)
        a[k] = *reinterpret_cast<const v2f*>(pa + 4 * k);

    int rowS[8];
#pragma unroll
    for (int v = 0; v < 8; ++v) {
        int rr = v + 8 * half;
        rowS[v] = bucket[base + (rr < cnt ? rr : cnt - 1)];
    }
    const float* W = aw + (size_t)t * D * D;
    float sk = 1.0f / (1.0f + expf(-skip[t]));

    v8f cc[4];
    for (int n = 0; n < 4; ++n) {
        v8f c = {};
        const int col = n * 16 + m;
#pragma unroll
        for (int k = 0; k < 16; ++k) {
            v2f b;
            b.x = W[(4 * k + kb) * D + col];
            b.y = W[(4 * k + kb + 1) * D + col];
            c = wmma_f32(a[k], b, c);
        }
        cc[n] = c;
    }
    // all reads of this tile's rows are complete (held in a[]) -> safe to overwrite
    for (int n = 0; n < 4; ++n) {
        const int col = n * 16 + m;
#pragma unroll
        for (int v = 0; v < 8; ++v)
            io[(size_t)rowS[v] * D + col] = cc[n][v] * sk;
    }
}

extern "C" void kernel_launch(void* const* d_in, const int* in_sizes, int n_in,
                              void* d_out, int out_size, void* d_ws, size_t ws_size,
                              hipStream_t stream) {
    const float* h     = (const float*)d_in[0];
    const int*   adj   = (const int*)d_in[1];
    const int*   etype = (const int*)d_in[2];
    const int*   ntype = (const int*)d_in[3];
    // d_in[4] src_type, d_in[5] dst_type: unused by the reference math
    const float* kw   = (const float*)d_in[6];
    const float* qw   = (const float*)d_in[7];
    const float* vw   = (const float*)d_in[8];
    const float* aw   = (const float*)d_in[9];
    const float* rpri = (const float*)d_in[10];
    const float* ratt = (const float*)d_in[11];
    const float* rmsg = (const float*)d_in[12];
    const float* skip = (const float*)d_in[13];

    const int Nn = in_sizes[3];   // node count
    const int Ee = in_sizes[2];   // edge count

    float* w = (float*)d_ws;
    float* kn = w;              w += (size_t)Nn * D;
    float* qn = w;              w += (size_t)Nn * D;
    float* vn = w;              w += (size_t)Nn * D;
    float* sb = w;              w += Ee;
    int*   nb   = (int*)w;      w += Nn;
    int*   ebuf = (int*)w;      w += Ee;
    int*   toff = (int*)w;      w += TNUM + 1;
    int*   roff = (int*)w;      w += RNUM + 1;
    int*   tcur = (int*)w;      w += TNUM;
    int*   rcur = (int*)w;      w += RNUM;
    // contiguous zero-init region:
    float*    ssum = w;             w += Nn;
    unsigned* mmax = (unsigned*)w;  w += Nn;
    int*      tcnt = (int*)w;       w += TNUM;
    int*      rcnt = (int*)w;       w += RNUM;

    float* out = (float*)d_out;     // also the message accumulator (agg)

    long zelems = 2L * Nn + TNUM + RNUM;
    k_zero<<<(unsigned)((zelems + 255) / 256), 256, 0, stream>>>(ssum, zelems);
    k_zero<<<(unsigned)(((long)Nn * D + 255) / 256), 256, 0, stream>>>(out, (long)Nn * D);

    k_count<<<(Nn + 255) / 256, 256, 0, stream>>>(ntype, Nn, tcnt);
    k_count<<<(Ee + 255) / 256, 256, 0, stream>>>(etype, Ee, rcnt);
    k_scan<<<1, 1, 0, stream>>>(tcnt, toff, tcur, TNUM);
    k_scan<<<1, 1, 0, stream>>>(rcnt, roff, rcur, RNUM);
    k_scatter<<<(Nn + 255) / 256, 256, 0, stream>>>(ntype, Nn, tcur, nb);
    k_scatter<<<(Ee + 255) / 256, 256, 0, stream>>>(etype, Ee, rcur, ebuf);

    unsigned ntiles = (unsigned)((Nn + 15) / 16 + TNUM);
    unsigned etiles = (unsigned)((Ee + 15) / 16 + RNUM);
    k_node_proj<<<ntiles, 32, 0, stream>>>(h, nb, toff, kw, qw, vw, kn, qn, vn);
    k_edge_score<<<etiles, 32, 0, stream>>>(kn, qn, adj, Ee, ebuf, roff, ratt, rpri, sb, mmax);
    k_exp_sum<<<(Ee + 255) / 256, 256, 0, stream>>>(sb, mmax, ssum, adj, Ee);
    k_edge_msg<<<etiles, 32, 0, stream>>>(vn, adj, Ee, ebuf, roff, rmsg, sb, ssum, out);
    k_out<<<ntiles, 32, 0, stream>>>(nb, toff, aw, skip, out);
}